// LinearInterpolator_87548613361887
// MI455X (gfx1250) — compile-verified
//
#include <hip/hip_runtime.h>
#include <stdint.h>

#define N_KNOTS 16384
#define BLOCK 256
#define CHUNKS 4                              // float4 chunks per thread
#define SAMP_PER_BLOCK (BLOCK * 4 * CHUNKS)   // 4096 samples per block

#if defined(__gfx1250__)
typedef int v4i_ls __attribute__((vector_size(16)));
typedef __attribute__((address_space(1))) v4i_ls* as1_v4i;   // global
typedef __attribute__((address_space(3))) v4i_ls* as3_v4i;   // LDS
#endif

__global__ __launch_bounds__(BLOCK)
void interp_kernel(const float* __restrict__ x_samp,
                   const float* __restrict__ x_points,
                   const float* __restrict__ y_points,
                   float* __restrict__ out,
                   int n_total)
{
    __shared__ float s_x[N_KNOTS];
    __shared__ float s_y[N_KNOTS];

    const int tid = threadIdx.x;

    // ---- Stage knot tables (2 x 64KB) into LDS -------------------------
#if defined(__gfx1250__) && __has_builtin(__builtin_amdgcn_global_load_async_to_lds_b128)
    // CDNA5 async copy path: global -> LDS without touching VGPRs,
    // tracked by ASYNCcnt.
    for (int i = tid * 4; i < N_KNOTS; i += BLOCK * 4) {
        __builtin_amdgcn_global_load_async_to_lds_b128(
            (as1_v4i)(uintptr_t)(x_points + i),
            (as3_v4i)(uint32_t)(uintptr_t)(s_x + i),
            0, 0);
        __builtin_amdgcn_global_load_async_to_lds_b128(
            (as1_v4i)(uintptr_t)(y_points + i),
            (as3_v4i)(uint32_t)(uintptr_t)(s_y + i),
            0, 0);
    }
#if __has_builtin(__builtin_amdgcn_s_wait_asynccnt)
    __builtin_amdgcn_s_wait_asynccnt(0);
#else
    asm volatile("s_wait_asynccnt 0" ::: "memory");
#endif
#else
    // Fallback (also taken by the host compile pass): plain cooperative copy.
    for (int i = tid * 4; i < N_KNOTS; i += BLOCK * 4) {
        *(float4*)(s_x + i) = *(const float4*)(x_points + i);
        *(float4*)(s_y + i) = *(const float4*)(y_points + i);
    }
#endif
    __syncthreads();

    const long long base = (long long)blockIdx.x * SAMP_PER_BLOCK;

#pragma unroll
    for (int c = 0; c < CHUNKS; ++c) {
        long long idx = base + (long long)(c * BLOCK + tid) * 4;
        if (idx + 3 >= n_total) {
            // scalar tail (not hit with the harness sizes: 8388608 % 4096 == 0)
            for (int k = 0; k < 4; ++k) {
                long long j = idx + k;
                if (j >= n_total) break;
                float x = x_samp[j];
                int pos = 0;
#pragma unroll
                for (int sh = 13; sh >= 0; --sh) {
                    int cand = pos + (1 << sh);
                    pos = (s_x[cand] <= x) ? cand : pos;
                }
                float xi = s_x[pos], xf = s_x[pos + 1];
                float yi = s_y[pos], yf = s_y[pos + 1];
                float ds = (x - xi) / (xf - xi);
                out[j] = fmaf(ds, yf - yi, yi);
            }
            continue;
        }

        float4 xs = *(const float4*)(x_samp + idx);
        float x0 = xs.x, x1 = xs.y, x2 = xs.z, x3 = xs.w;

        // Branchless binary search: 4 independent dependent-load chains
        // interleaved so LDS latency overlaps.
        int p0 = 0, p1 = 0, p2 = 0, p3 = 0;
#pragma unroll
        for (int sh = 13; sh >= 0; --sh) {
            const int step = 1 << sh;
            int c0 = p0 + step, c1 = p1 + step, c2 = p2 + step, c3 = p3 + step;
            float v0 = s_x[c0], v1 = s_x[c1], v2 = s_x[c2], v3 = s_x[c3];
            p0 = (v0 <= x0) ? c0 : p0;
            p1 = (v1 <= x1) ? c1 : p1;
            p2 = (v2 <= x2) ? c2 : p2;
            p3 = (v3 <= x3) ? c3 : p3;
        }

        float xi0 = s_x[p0], xf0 = s_x[p0 + 1], yi0 = s_y[p0], yf0 = s_y[p0 + 1];
        float xi1 = s_x[p1], xf1 = s_x[p1 + 1], yi1 = s_y[p1], yf1 = s_y[p1 + 1];
        float xi2 = s_x[p2], xf2 = s_x[p2 + 1], yi2 = s_y[p2], yf2 = s_y[p2 + 1];
        float xi3 = s_x[p3], xf3 = s_x[p3 + 1], yi3 = s_y[p3], yf3 = s_y[p3 + 1];

        float4 r;
        r.x = fmaf((x0 - xi0) / (xf0 - xi0), yf0 - yi0, yi0);
        r.y = fmaf((x1 - xi1) / (xf1 - xi1), yf1 - yi1, yi1);
        r.z = fmaf((x2 - xi2) / (xf2 - xi2), yf2 - yi2, yi2);
        r.w = fmaf((x3 - xi3) / (xf3 - xi3), yf3 - yi3, yi3);

        *(float4*)(out + idx) = r;
    }
}

extern "C" void kernel_launch(void* const* d_in, const int* in_sizes, int n_in,
                              void* d_out, int out_size, void* d_ws, size_t ws_size,
                              hipStream_t stream)
{
    const float* x_samp   = (const float*)d_in[0];
    const float* x_points = (const float*)d_in[1];
    const float* y_points = (const float*)d_in[2];
    float* out = (float*)d_out;

    const int n = in_sizes[0];  // BATCH * N_SAMP = 8,388,608
    const int grid = (n + SAMP_PER_BLOCK - 1) / SAMP_PER_BLOCK;  // 2048

    hipLaunchKernelGGL(interp_kernel, dim3(grid), dim3(BLOCK), 0, stream,
                       x_samp, x_points, y_points, out, n);
}